// USChannelDrop_28613072126356
// MI455X (gfx1250) — compile-verified
//
#include <hip/hip_runtime.h>
#include <hip/hip_bf16.h>

// Problem constants from the reference: x = (64, 256, 56, 56) fp32
#define NBATCH   64
#define NCHAN    256
#define PLANE    3136            // 56*56 floats per (b,c) plane = 12544 bytes
#define NPLANES  (NBATCH * NCHAN)   // 16384
#define KCHAN    128
#define NP       4               // planes per workgroup in the reduction pass

typedef unsigned int v4u __attribute__((ext_vector_type(4)));
typedef int          v4i __attribute__((ext_vector_type(4)));
typedef int          v8i __attribute__((ext_vector_type(8)));
typedef float        v2f __attribute__((ext_vector_type(2)));
typedef float        v4f __attribute__((ext_vector_type(4)));

// ---------------------------------------------------------------------------
// TDM: issue one tensor_load_to_lds moving a full 3136-float plane into LDS.
// D# packing per CDNA5 ISA ch. 8 (group0: count/lds_addr/global_addr/type,
// group1: data_size=4B, tensor_dim0=3136, tile_dim0=3136, 1 row).
// Tracked by TENSORcnt; EXEC-independent (one op per wave).
// ---------------------------------------------------------------------------
__device__ __forceinline__ void tdm_load_plane(const float* gsrc, unsigned lds_off) {
    unsigned long long ga = (unsigned long long)(size_t)gsrc;

    v4u g0;
    g0.x = 1u;                                    // [1:0] count=1 (valid user D#)
    g0.y = lds_off;                               // [63:32] lds_addr (bytes)
    g0.z = (unsigned)(ga & 0xffffffffu);          // [95:64] global_addr lo
    g0.w = (unsigned)((ga >> 32) & 0x01ffffffu)   // [120:96] global_addr hi
         | (2u << 30);                            // [127:126] type = 2 ("image")

    v8i g1;
    g1[0] = (int)(2u << 16);                      // data_size=2 (4 bytes); wg_mask=0
    g1[1] = (int)(((unsigned)PLANE & 0xffffu) << 16); // tensor_dim0[15:0] @ bits 63:48
    g1[2] = (int)(((unsigned)PLANE >> 16)             // tensor_dim0[31:16]
         | (1u << 16));                           // tensor_dim1 = 1
    g1[3] = (int)(((unsigned)PLANE) << 16);       // tile_dim0 = 3136 @ bits 127:112
    g1[4] = 1;                                    // tile_dim1 = 1, tile_dim2 = 0
    g1[5] = PLANE;                                // tensor_dim0_stride lo32
    g1[6] = (int)(((unsigned)PLANE) << 16);       // stride0 hi16=0 | tensor_dim1_stride lo16
    g1[7] = 0;                                    // tensor_dim1_stride hi32

    v4i z4 = {0, 0, 0, 0};
#if defined(__clang_major__) && (__clang_major__ >= 23)
    v8i z8 = {0, 0, 0, 0, 0, 0, 0, 0};
    __builtin_amdgcn_tensor_load_to_lds(g0, g1, z4, z4, z8, 0);
#else
    __builtin_amdgcn_tensor_load_to_lds(g0, g1, z4, z4, 0);
#endif
}

// ---------------------------------------------------------------------------
// Pass 1: per-(b,c) sum of squares. One wave (32 lanes) per workgroup,
// NP planes per workgroup, double-buffered TDM loads into LDS.
// ---------------------------------------------------------------------------
__global__ __launch_bounds__(32)
void uscd_mag_kernel(const float* __restrict__ x, float* __restrict__ mag) {
    __shared__ float buf[2][PLANE];               // 2 x 12544 B ping-pong
    const int lane = threadIdx.x;                 // 0..31 (wave32)
    const int base = blockIdx.x * NP;

    const unsigned lds0 = (unsigned)(size_t)(void*)&buf[0][0];
    const unsigned lds1 = (unsigned)(size_t)(void*)&buf[1][0];

    tdm_load_plane(x + (size_t)base * PLANE, lds0);

    for (int p = 0; p < NP; ++p) {
        if (p + 1 < NP) {
            tdm_load_plane(x + (size_t)(base + p + 1) * PLANE,
                           ((p + 1) & 1) ? lds1 : lds0);
            __builtin_amdgcn_s_wait_tensorcnt(1); // plane p landed; p+1 in flight
        } else {
            __builtin_amdgcn_s_wait_tensorcnt(0); // drain
        }

        const v2f* src = (const v2f*)((p & 1) ? &buf[1][0] : &buf[0][0]);
        float s = 0.0f;
        #pragma unroll
        for (int k = 0; k < PLANE / 64; ++k) {    // 49 x ds_load_b64 per lane
            v2f d = src[k * 32 + lane];
            s = fmaf(d.x, d.x, s);
            s = fmaf(d.y, d.y, s);
        }
        // wave32 butterfly reduction
        for (int off = 16; off > 0; off >>= 1)
            s += __shfl_xor(s, off, 32);
        if (lane == 0) mag[base + p] = s;
    }
}

// ---------------------------------------------------------------------------
// Pass 2: per-batch 128-th largest of 256 energies (order statistic by
// counting; tie-safe: all candidates hold the identical value).
// ---------------------------------------------------------------------------
__global__ __launch_bounds__(NCHAN)
void uscd_thresh_kernel(const float* __restrict__ mag, float* __restrict__ thr) {
    __shared__ float m[NCHAN];
    const int b = blockIdx.x;
    const int c = threadIdx.x;
    const float v = mag[b * NCHAN + c];
    m[c] = v;
    __syncthreads();

    int gt = 0, ge = 0;
    #pragma unroll 8
    for (int j = 0; j < NCHAN; ++j) {
        float o = m[j];
        gt += (o >  v);
        ge += (o >= v);
    }
    if (gt <= KCHAN - 1 && ge >= KCHAN) thr[b] = v;   // v == sorted_desc[K-1]
}

// ---------------------------------------------------------------------------
// Pass 3: y = (mag > thr) ? x : 0. Dropped channels are pure zero-stores (no
// read). Non-temporal b128 traffic; reverse plane order to catch the tail of
// x in the 192 MB L2 left warm by pass 1.
// ---------------------------------------------------------------------------
__global__ __launch_bounds__(256)
void uscd_apply_kernel(const float* __restrict__ x,
                       const float* __restrict__ mag,
                       const float* __restrict__ thr,
                       float* __restrict__ y) {
    const int plane = NPLANES - 1 - blockIdx.x;   // reverse order for L2 reuse
    const int b = plane >> 8;                     // plane / NCHAN
    const bool keep = mag[plane] > thr[b];

    const v4f* xin  = (const v4f*)(x + (size_t)plane * PLANE);
    v4f*       yout = (v4f*)(y + (size_t)plane * PLANE);

    if (keep) {
        for (int i = threadIdx.x; i < PLANE / 4; i += 256) {
            v4f v = __builtin_nontemporal_load(xin + i);
            __builtin_nontemporal_store(v, yout + i);
        }
    } else {
        const v4f z = {0.0f, 0.0f, 0.0f, 0.0f};
        for (int i = threadIdx.x; i < PLANE / 4; i += 256)
            __builtin_nontemporal_store(z, yout + i);
    }
}

// ---------------------------------------------------------------------------
extern "C" void kernel_launch(void* const* d_in, const int* in_sizes, int n_in,
                              void* d_out, int out_size, void* d_ws, size_t ws_size,
                              hipStream_t stream) {
    (void)in_sizes; (void)n_in; (void)out_size; (void)ws_size;
    const float* x = (const float*)d_in[0];
    float* y   = (float*)d_out;
    float* mag = (float*)d_ws;            // 16384 floats
    float* thr = mag + NPLANES;           // 64 floats

    uscd_mag_kernel   <<<NPLANES / NP, 32,  0, stream>>>(x, mag);
    uscd_thresh_kernel<<<NBATCH,       NCHAN, 0, stream>>>(mag, thr);
    uscd_apply_kernel <<<NPLANES,      256, 0, stream>>>(x, mag, thr, y);
}